// MultiGranularityDecomposer_34024730919670
// MI455X (gfx1250) — compile-verified
//
#include <hip/hip_runtime.h>

#define T_LEN   600
#define PERIOD  144
#define ROWS_PB 8
#define CHUNK   19   // ceil(600/32)

typedef __attribute__((ext_vector_type(2))) float v2f;
typedef __attribute__((ext_vector_type(4))) float f4;
typedef __attribute__((ext_vector_type(8))) float v8f;
typedef int v4i __attribute__((vector_size(16)));   // matches builtin param type

#define AS1 __attribute__((address_space(1)))
#define AS3 __attribute__((address_space(3)))

#if __has_builtin(__builtin_amdgcn_global_load_async_to_lds_b128)
#define HAVE_ASYNC_LDS 1
#else
#define HAVE_ASYNC_LDS 0
#endif

#if __has_builtin(__builtin_amdgcn_s_wait_asynccnt)
#warning "ATHENA: s_wait_asynccnt builtin AVAILABLE"
#else
#warning "ATHENA: s_wait_asynccnt builtin MISSING (using inline asm)"
#endif

struct __align__(16) RowBuf {
  float x[600];   // staged input row (2400 B, 16B aligned)
  float S[604];   // prefix sums S[0..600]
  float mk[148];  // 144 phase means
};

__global__ __launch_bounds__(256) void mgd_kernel(
    const float* __restrict__ xg,
    float* __restrict__ hf_o,
    float* __restrict__ dy_o,
    float* __restrict__ tr_o,
    long long nrows)
{
  __shared__ RowBuf rb[ROWS_PB];
  const int lane = threadIdx.x & 31;
  const int w    = threadIdx.x >> 5;
  const long long row = (long long)blockIdx.x * ROWS_PB + w;
  if (row >= nrows) return;   // wave-uniform

  float* xs = rb[w].x;
  float* Sp = rb[w].S;
  float* mk = rb[w].mk;
  const float* src = xg + row * (long long)T_LEN;

  // ---- stage input row into LDS (CDNA5 async global->LDS path) ------------
#if HAVE_ASYNC_LDS
  {
    unsigned long long g = (unsigned long long)src;
    unsigned l = (unsigned)(unsigned long long)xs;
    for (int c = 0; c < 5; ++c) {
      const int bo = c * 512 + lane * 16;     // 2400 = 4*512 + 22*16
      if (bo < 2400) {
        __builtin_amdgcn_global_load_async_to_lds_b128(
            (AS1 v4i*)(g + (unsigned long long)bo),
            (AS3 v4i*)(l + (unsigned)bo),
            /*imm offset*/0, /*cpol*/0);
      }
    }
  }
#if __has_builtin(__builtin_amdgcn_s_wait_asynccnt)
  __builtin_amdgcn_s_wait_asynccnt(0);
#else
  asm volatile("s_wait_asynccnt 0" ::: "memory");
#endif
#else
  for (int i = lane; i < 150; i += 32) {
    f4 v = __builtin_nontemporal_load(((const f4*)src) + i);
    ((f4*)xs)[i] = v;
  }
#endif
  __builtin_amdgcn_wave_barrier();
  asm volatile("" ::: "memory");

  // ---- daily phase means over the 4 full periods --------------------------
  // mean_k[p] = 0.25*(x[p] + x[p+144] + x[p+288] + x[p+432]), p in [0,144)
#if __has_builtin(__builtin_amdgcn_wmma_f32_16x16x4_f32)
  {
    const int  l16 = lane & 15;
    const bool hiH = lane >= 16;
    v2f A; A[0] = 0.25f; A[1] = 0.25f;   // A[m,k] = 1/4 for all m,k
    #pragma unroll
    for (int g9 = 0; g9 < 9; ++g9) {     // 9 * 16 = 144 phases
      const int p0 = g9 * 16;
      // B (4x16): V0 = {K0 | K2}, V1 = {K1 | K3}; lane%16 = N (phase)
      v2f B;
      B[0] = xs[p0 + l16 + (hiH ? 288 : 0)];
      B[1] = xs[p0 + l16 + (hiH ? 432 : 144)];
      v8f C = {};
      v8f D = __builtin_amdgcn_wmma_f32_16x16x4_f32(
          false, A, false, B, (short)0, C, false, false);
      // every row of C equals the column means; row 0 lives in D[0], lanes 0-15
      if (lane < 16) mk[p0 + l16] = D[0];
    }
  }
#else
  for (int p = lane; p < PERIOD; p += 32)
    mk[p] = 0.25f * (xs[p] + xs[p + 144] + xs[p + 288] + xs[p + 432]);
#endif

  // ---- prefix sums for the K=25 moving average ----------------------------
  {
    const int t0 = lane * CHUNK;
    const int t1 = (t0 + CHUNK < T_LEN) ? (t0 + CHUNK) : T_LEN;
    float part = 0.f;
    for (int t = t0; t < t1; ++t) part += xs[t];
    float scan = part;
    #pragma unroll
    for (int d = 1; d < 32; d <<= 1) {
      float nv = __shfl_up(scan, d, 32);
      if (lane >= d) scan += nv;
    }
    float run = scan - part;              // exclusive prefix of lane chunks
    if (lane == 0) Sp[0] = 0.f;
    for (int t = t0; t < t1; ++t) { run += xs[t]; Sp[t + 1] = run; }
  }
  __builtin_amdgcn_wave_barrier();
  asm volatile("" ::: "memory");

  // ---- emit hf / daily / trend with coalesced float4 NT stores ------------
  {
    const long long ob = row * (long long)T_LEN;
    f4* hf4 = (f4*)(hf_o + ob);
    f4* dy4 = (f4*)(dy_o + ob);
    f4* tr4 = (f4*)(tr_o + ob);
    #pragma unroll
    for (int i = 0; i < 5; ++i) {
      const int q = lane + i * 32;        // float4 index, 150 per row
      if (q < 150) {
        const int tb = q * 4;
        f4 trv, dyv, hfv;
        #pragma unroll
        for (int j = 0; j < 4; ++j) {
          const int t = tb + j;
          int lo = t - 12; lo = lo < 0 ? 0 : lo;
          int hi = t + 13; hi = hi > T_LEN ? T_LEN : hi;
          const float tr = (Sp[hi] - Sp[lo]) * 0.04f;   // count_include_pad: /25
          const float dy = mk[t % PERIOD];              // 576 = 4*144 -> tail == mod
          const float lw = 0.5f * tr + 0.5f * dy;
          trv[j] = tr;
          dyv[j] = dy;
          hfv[j] = xs[t] - lw;
        }
        __builtin_nontemporal_store(hfv, hf4 + q);
        __builtin_nontemporal_store(dyv, dy4 + q);
        __builtin_nontemporal_store(trv, tr4 + q);
      }
    }
  }
}

extern "C" void kernel_launch(void* const* d_in, const int* in_sizes, int n_in,
                              void* d_out, int out_size, void* d_ws, size_t ws_size,
                              hipStream_t stream) {
  const float* x = (const float*)d_in[0];
  const long long n = (long long)in_sizes[0];   // 8*32*207*600 = 31,795,200
  const long long nrows = n / T_LEN;            // 52,992 rows
  float* out = (float*)d_out;
  float* hf = out;                              // return order: hf, daily, trend
  float* dy = out + n;
  float* tr = out + 2 * n;
  dim3 block(256);
  dim3 grid((unsigned)((nrows + ROWS_PB - 1) / ROWS_PB));
  mgd_kernel<<<grid, block, 0, stream>>>(x, hf, dy, tr, nrows);
}